// NeuroNN_9603546874014
// MI455X (gfx1250) — compile-verified
//
#include <hip/hip_runtime.h>
#include <hip/hip_bf16.h>

typedef float v2f __attribute__((ext_vector_type(2)));
typedef float v8f __attribute__((ext_vector_type(8)));

#define LD     1504     // padded N (94 * 16)
#define NN     1500
#define NE_    1200
#define MB     96       // 8 contrasts * 12 orients
#define ITILES 94
#define BTILES 6
#define D2R    0.017453292519943295f

// preferred orientation of neuron k (linspace endpoint=False over [0,179.99))
__device__ __forceinline__ float pref_of(int k) {
    return (k < NE_) ? (179.99f * (float)k * (1.0f / 1200.0f))
                     : (179.99f * (float)(k - NE_) * (1.0f / 300.0f));
}

// ---------------- Ricciardi transfer pieces ----------------
__device__ __forceinline__ float f_ricci(float x) {
    float z = x / (1.0f + x);
    float t = -z;
    float p = 0.14805913578876898f;
    p = p * t + 0.64290613877355551f;
    p = p * t + 1.0616084849547165f;
    p = p * t + 0.93524391761244940f;
    p = p * t + 0.62718906618071668f;
    p = p * t + 0.32171431660633076f;
    p = p * t + 0.32056016125642045f;
    p = p * t + 0.77373949685442023f;
    p = p * t + 0.22757881388024176f;
    p = p * t;                      // a0 = 0
    return logf(2.0f * x + 1.0f) + p;
}

__device__ __forceinline__ float g_ricci(float x) {
    float z = x / (2.0f + x);
    float num = z * (3.5441754117462949f + z * (-7.0529131065835378f + z * (-56.532378057580381f
              + z * (279.56761105465944f + z * (-520.37554849441472f + z * (456.58245777026514f
              + z * (-155.73340457809226f)))))));
    float den = 1.0f + z * (-4.1357968834226053f + z * (-7.2984226138266743f + z * (98.656602235468327f
              + z * (-334.20436223415163f + z * (601.08633903294185f + z * (-599.58577549598340f
              + z * (277.18420330693891f + z * (-16.445022798669722f))))))));
    return num / den;
}

__device__ __forceinline__ float phi_f(float mu, float sig, float tau, float tau_ref) {
    float xp = mu / sig;               // VR = 0
    float xm = (mu - 20.0f) / sig;     // VT = 20
    float r0;
    if (xm > 0.0f) {
        r0 = 1.0f / (f_ricci(xp) - f_ricci(xm));
    } else if (xp > 0.0f) {
        r0 = 1.0f / (f_ricci(xp) + expf(xm * xm) * g_ricci(-xm));
    } else {
        r0 = expf(-xm * xm - logf(g_ricci(-xm) - expf(xp * xp - xm * xm) * g_ricci(-xp)));
    }
    r0 = fmaxf(r0, 1e-30f);
    return 1.0f / (tau_ref + tau / r0);
}

// ---------------- one-time setup kernels ----------------
__global__ void init_w_kernel(const float* __restrict__ hyp, const float* __restrict__ rnd,
                              float* __restrict__ W, float* __restrict__ W2) {
    int j = blockIdx.x * blockDim.x + threadIdx.x;
    int i = blockIdx.y;
    if (j >= LD) return;
    float w = 0.0f;
    if (i < NN && j < NN) {
        int conn = ((i >= NE_) ? 1 : 0) + ((j >= NE_) ? 2 : 0);
        float J  = hyp[conn];
        float P  = hyp[4 + conn];
        float Wc = hyp[8 + conn];
        float diff  = fabsf(pref_of(i) - pref_of(j));
        float denom = 4.0f * (D2R * Wc) * (D2R * Wc);
        float z = expf((cosf(2.0f * D2R * diff) - 1.0f) / denom);
        float s = 32.0f * (P * z - rnd[i * NN + j]);
        w = J / (1.0f + expf(-s));
    }
    W[i * LD + j]  = w;
    W2[i * LD + j] = w * w;
}

__global__ void init_mean_kernel(float* __restrict__ mean, float* __restrict__ rate0) {
    int n = blockIdx.x * blockDim.x + threadIdx.x;
    int b = blockIdx.y;
    if (n >= LD) return;
    const float contrasts[8] = {0.0f, 0.0432773f, 0.103411f, 0.186966f,
                                0.303066f, 0.464386f, 0.68854f, 1.0f};
    float v = 0.0f;
    if (n < NN) {
        int c = b / 12, o = b % 12;
        float orient = 15.0f * (float)o;
        float dth = orient - pref_of(n);
        float denom = 4.0f * (D2R * 30.0f) * (D2R * 30.0f);   // W_FF = 30
        float cg = expf((cosf(2.0f * D2R * dth) - 1.0f) / denom);
        v = contrasts[c] * 20.0f * cg;                        // SCALING_G = 1
    }
    mean[b * LD + n]  = v;
    rate0[b * LD + n] = 0.0f;
}

// ---------------- one Euler step: 2 GEMMs (f32 WMMA) + phi ----------------
__global__ __launch_bounds__(128) void step_kernel(
        const float* __restrict__ W, const float* __restrict__ W2,
        const float* __restrict__ mean, const float* __restrict__ rin,
        float* __restrict__ rout) {
    int gwave = (int)(blockIdx.x * 128 + threadIdx.x) >> 5;   // 0..563, exact fit
    int lane  = threadIdx.x & 31;
    int btile = gwave % BTILES;
    int itile = gwave / BTILES;
    int b0 = btile * 16;
    int i0 = itile * 16;

    int half = lane >> 4;          // 0: K={0,1}, 1: K={2,3} (A/B f32 layouts)
    int l15  = lane & 15;          // A: M row; B: N column
    int koff = half * 2;

    const float* arow  = rin + (size_t)(b0 + l15) * LD + koff;
    const float* w1row = W   + (size_t)(i0 + l15) * LD + koff;
    const float* w2row = W2  + (size_t)(i0 + l15) * LD + koff;

    v8f acc1 = {0.f, 0.f, 0.f, 0.f, 0.f, 0.f, 0.f, 0.f};
    v8f acc2 = {0.f, 0.f, 0.f, 0.f, 0.f, 0.f, 0.f, 0.f};

#pragma unroll 4
    for (int k = 0; k < LD; k += 4) {
        float2 af  = *(const float2*)(arow  + k);
        float2 b1f = *(const float2*)(w1row + k);
        float2 b2f = *(const float2*)(w2row + k);
        v2f A  = {af.x,  af.y};
        v2f B1 = {b1f.x, b1f.y};
        v2f B2 = {b2f.x, b2f.y};
        // D = A(16x4 f32) x B(4x16 f32) + C, full fp32 accumulate
        acc1 = __builtin_amdgcn_wmma_f32_16x16x4_f32(false, A, false, B1,
                                                     (short)0, acc1, false, false);
        acc2 = __builtin_amdgcn_wmma_f32_16x16x4_f32(false, A, false, B2,
                                                     (short)0, acc2, false, false);
    }

    int n = i0 + l15;
    const float tau = 0.01f;
    float tinv = (n < NE_) ? 100.0f : 200.0f;
    float tref = (n < NE_) ? 0.005f : 0.001f;
    bool  vn   = (n < NN);

#pragma unroll
    for (int r = 0; r < 8; ++r) {
        int b = b0 + r + half * 8;                 // C/D layout: M = r + 8*(lane>=16)
        float mu = tau * acc1[r] + mean[(size_t)b * LD + n];
        float sg = sqrtf(tau * acc2[r] + 25.0f);   // SIG_EXT^2 = 25
        float ro = rin[(size_t)b * LD + n];
        float ph = phi_f(mu, sg, tau, tref);
        float rn = ro + 0.001f * tinv * (ph - ro); // DT = 1e-3
        rout[(size_t)b * LD + n] = vn ? rn : 0.0f;
    }
}

// ---------------- final transpose [96,N] -> [N,8,12] ----------------
__global__ void out_kernel(const float* __restrict__ rate, float* __restrict__ out) {
    int idx = blockIdx.x * blockDim.x + threadIdx.x;
    if (idx >= NN * MB) return;
    int nrow = idx / MB;
    int b    = idx % MB;
    out[idx] = rate[(size_t)b * LD + nrow];
}

extern "C" void kernel_launch(void* const* d_in, const int* in_sizes, int n_in,
                              void* d_out, int out_size, void* d_ws, size_t ws_size,
                              hipStream_t stream) {
    const float* hyp = (const float*)d_in[0];   // [3,4] row-major
    const float* rnd = (const float*)d_in[1];   // [1500,1500]
    (void)in_sizes; (void)n_in; (void)out_size; (void)ws_size;

    float* ws   = (float*)d_ws;
    float* W    = ws;
    float* W2   = W    + (size_t)LD * LD;
    float* mean = W2   + (size_t)LD * LD;
    float* rA   = mean + (size_t)MB * LD;
    float* rB   = rA   + (size_t)MB * LD;

    dim3 gw((LD + 255) / 256, LD);
    init_w_kernel<<<gw, 256, 0, stream>>>(hyp, rnd, W, W2);

    dim3 gm((LD + 255) / 256, MB);
    init_mean_kernel<<<gm, 256, 0, stream>>>(mean, rA);

    const int nblocks = (BTILES * ITILES * 32 + 127) / 128;   // 141 blocks, 4 waves each
    float* cur = rA;
    float* nxt = rB;
    for (int it = 0; it < 60; ++it) {                         // NMAX = 60
        step_kernel<<<nblocks, 128, 0, stream>>>(W, W2, mean, cur, nxt);
        float* t = cur; cur = nxt; nxt = t;
    }

    out_kernel<<<(NN * MB + 255) / 256, 256, 0, stream>>>(cur, (float*)d_out);
}